// CausalHAFE_Baseline_5523327942985
// MI455X (gfx1250) — compile-verified
//
#include <hip/hip_runtime.h>

// ---------------------------------------------------------------------------
// Type-aware GCN, restructured for CDNA5 (gfx1250):
//   S[n,t,:] = sum over edges(dst=n, type=t) of x[src]        (f32 atomics)
//   h        = [S * (1/deg) | x] @ [W_type ; W_self] + b      (V_WMMA_F32_16X16X4_F32)
// ---------------------------------------------------------------------------

typedef __attribute__((ext_vector_type(2))) float v2f;
typedef __attribute__((ext_vector_type(8))) float v8f;

#define H_DIM   256
#define T_TYPES 4

// ---- degree accumulation (one thread per edge) ----------------------------
__global__ void deg_kernel(const int* __restrict__ dst, float* __restrict__ deg, int nE) {
    int e = blockIdx.x * blockDim.x + threadIdx.x;
    if (e < nE) atomicAdd(&deg[dst[e]], 1.0f);
}

// ---- deg -> 1/max(deg,1) in place ----------------------------------------
__global__ void rdeg_kernel(float* __restrict__ deg, int n) {
    int i = blockIdx.x * blockDim.x + threadIdx.x;
    if (i < n) deg[i] = 1.0f / fmaxf(deg[i], 1.0f);
}

// ---- per-(dst,type) feature scatter-sum (one wave32 per edge) -------------
__global__ void scatter_kernel(const float* __restrict__ x, float* __restrict__ S,
                               const int* __restrict__ src, const int* __restrict__ dst,
                               const int* __restrict__ et, int nE, int D) {
    int wid  = blockIdx.x * (blockDim.x >> 5) + (threadIdx.x >> 5);
    int lane = threadIdx.x & 31;
    if (wid >= nE) return;
    int s = src[wid], d = dst[wid], t = et[wid];
    const float* xs = x + (size_t)s * D;
    float*       Sd = S + ((size_t)d * T_TYPES + t) * D;
    for (int i = lane; i < D; i += 32)
        atomicAdd(&Sd[i], xs[i]);
}

// ---- stacked GEMM with fused degree normalization, bias, (relu) -----------
// A (per row, Ktot = Kt + Kin cols) = [ S[row,:]*rdeg[row] | X[row,:] ]
// B (Ktot x 256)                    = [ W_type (flattened) ; W_self ]
// Block = 4 waves (128 thr); wave -> 32 rows x 64 cols (2 row tiles x 4 col
// tiles, 8 v8f accumulators); block -> 32 rows x 256 cols.
__global__ void __launch_bounds__(128)
gemm_kernel(const float* __restrict__ S, const float* __restrict__ X,
            const float* __restrict__ Wt, const float* __restrict__ Ws,
            const float* __restrict__ bias, const float* __restrict__ rdeg,
            float* __restrict__ out, int Kt, int Kin, int doRelu) {
    const int lane     = threadIdx.x & 31;
    const int wave     = threadIdx.x >> 5;
    const int rowsBase = blockIdx.x * 32;
    const int c0       = wave * 64;
    const int m        = lane & 15;
    const int aHalf    = (lane >> 4) << 1;            // K sub-offset: 0 or 2
    const int row0     = rowsBase + m;
    const int row1     = row0 + 16;
    const float rs0    = rdeg[row0];
    const float rs1    = rdeg[row1];

    v8f acc0[4] = {};     // rows rowsBase .. rowsBase+15
    v8f acc1[4] = {};     // rows rowsBase+16 .. rowsBase+31

    // ---- part 1: degree-scaled neighbor sums vs W_type --------------------
    {
        const float* S0 = S + (size_t)row0 * Kt + aHalf;
        const float* S1 = S + (size_t)row1 * Kt + aHalf;
        for (int kb = 0; kb < Kt; kb += 4) {
            v2f a0 = *(const v2f*)(S0 + kb);  a0 *= rs0;
            v2f a1 = *(const v2f*)(S1 + kb);  a1 *= rs1;
            const float* b0p = Wt + (size_t)(kb + aHalf) * H_DIM;
#pragma unroll
            for (int j = 0; j < 4; ++j) {
                const int n = c0 + j * 16 + m;
                v2f b;
                b.x = b0p[n];
                b.y = b0p[H_DIM + n];
                acc0[j] = __builtin_amdgcn_wmma_f32_16x16x4_f32(
                    false, a0, false, b, (short)0, acc0[j], false, false);
                acc1[j] = __builtin_amdgcn_wmma_f32_16x16x4_f32(
                    false, a1, false, b, (short)0, acc1[j], false, false);
            }
        }
    }

    // ---- part 2: self features vs W_self ----------------------------------
    {
        const float* X0 = X + (size_t)row0 * Kin + aHalf;
        const float* X1 = X + (size_t)row1 * Kin + aHalf;
        for (int kb = 0; kb < Kin; kb += 4) {
            v2f a0 = *(const v2f*)(X0 + kb);
            v2f a1 = *(const v2f*)(X1 + kb);
            const float* b0p = Ws + (size_t)(kb + aHalf) * H_DIM;
#pragma unroll
            for (int j = 0; j < 4; ++j) {
                const int n = c0 + j * 16 + m;
                v2f b;
                b.x = b0p[n];
                b.y = b0p[H_DIM + n];
                acc0[j] = __builtin_amdgcn_wmma_f32_16x16x4_f32(
                    false, a0, false, b, (short)0, acc0[j], false, false);
                acc1[j] = __builtin_amdgcn_wmma_f32_16x16x4_f32(
                    false, a1, false, b, (short)0, acc1[j], false, false);
            }
        }
    }

    // ---- epilogue: C VGPR r -> row = tileBase + r (+8 for lanes 16..31) ---
#pragma unroll
    for (int j = 0; j < 4; ++j) {
        const int col  = c0 + j * 16 + m;
        const float bv = bias[col];
        const int rA   = rowsBase + ((lane >> 4) << 3);
        const int rB   = rA + 16;
#pragma unroll
        for (int r = 0; r < 8; ++r) {
            float v0 = acc0[j][r] + bv;
            float v1 = acc1[j][r] + bv;
            if (doRelu) { v0 = fmaxf(v0, 0.0f); v1 = fmaxf(v1, 0.0f); }
            out[(size_t)(rA + r) * H_DIM + col] = v0;
            out[(size_t)(rB + r) * H_DIM + col] = v1;
        }
    }
}

// ---- classifier over gathered aspect nodes (tiny) -------------------------
__global__ void cls_kernel(const float* __restrict__ h, const int* __restrict__ aspect,
                           const float* __restrict__ Wc, const float* __restrict__ bc,
                           float* __restrict__ out, int nA, int nC) {
    int tid = blockIdx.x * blockDim.x + threadIdx.x;
    if (tid >= nA * nC) return;
    int a = tid / nC, c = tid % nC;
    const float* hv = h + (size_t)aspect[a] * H_DIM;
    float s = bc[c];
    for (int k = 0; k < H_DIM; ++k)
        s += hv[k] * Wc[k * nC + c];
    out[tid] = s;
}

extern "C" void kernel_launch(void* const* d_in, const int* in_sizes, int n_in,
                              void* d_out, int out_size, void* d_ws, size_t ws_size,
                              hipStream_t stream) {
    const float* features = (const float*)d_in[0];
    const float* W_type1  = (const float*)d_in[1];
    const float* W_self1  = (const float*)d_in[2];
    const float* b1       = (const float*)d_in[3];
    const float* W_type2  = (const float*)d_in[4];
    const float* W_self2  = (const float*)d_in[5];
    const float* b2       = (const float*)d_in[6];
    const float* W_cls    = (const float*)d_in[7];
    const float* b_cls    = (const float*)d_in[8];
    const int*   edge_idx = (const int*)d_in[9];
    const int*   etypes   = (const int*)d_in[10];
    const int*   aspects  = (const int*)d_in[11];

    const int Kin1 = in_sizes[2] / H_DIM;       // 128
    const int nN   = in_sizes[0] / Kin1;        // 20000
    const int nE   = in_sizes[10];              // 640000
    const int nA   = in_sizes[11];              // 4096
    const int nC   = in_sizes[8];               // 3

    const int* src = edge_idx;
    const int* dst = edge_idx + nE;

    // Workspace layout: deg | h1 | h2 | S (S reused for both layers)
    char* w = (char*)d_ws;
    size_t off = 0;
    float* deg = (float*)(w + off); off += (((size_t)nN * 4) + 255) & ~(size_t)255;
    float* h1  = (float*)(w + off); off += (size_t)nN * H_DIM * 4;
    float* h2  = (float*)(w + off); off += (size_t)nN * H_DIM * 4;
    float* Sb  = (float*)(w + off);
    const size_t S1bytes = (size_t)nN * T_TYPES * Kin1  * 4;   // 40.96 MB
    const size_t S2bytes = (size_t)nN * T_TYPES * H_DIM * 4;   // 81.92 MB

    const int Kt1 = T_TYPES * Kin1;    // 512
    const int Kt2 = T_TYPES * H_DIM;   // 1024

    // ---- degrees ----
    hipMemsetAsync(deg, 0, (size_t)nN * 4, stream);
    deg_kernel<<<(nE + 255) / 256, 256, 0, stream>>>(dst, deg, nE);
    rdeg_kernel<<<(nN + 255) / 256, 256, 0, stream>>>(deg, nN);

    // ---- layer 1 ----
    hipMemsetAsync(Sb, 0, S1bytes, stream);
    scatter_kernel<<<(nE + 7) / 8, 256, 0, stream>>>(features, Sb, src, dst, etypes, nE, Kin1);
    gemm_kernel<<<nN / 32, 128, 0, stream>>>(Sb, features, W_type1, W_self1, b1, deg,
                                             h1, Kt1, Kin1, /*relu=*/1);

    // ---- layer 2 ----
    hipMemsetAsync(Sb, 0, S2bytes, stream);
    scatter_kernel<<<(nE + 7) / 8, 256, 0, stream>>>(h1, Sb, src, dst, etypes, nE, H_DIM);
    gemm_kernel<<<nN / 32, 128, 0, stream>>>(Sb, h1, W_type2, W_self2, b2, deg,
                                             h2, Kt2, H_DIM, /*relu=*/0);

    // ---- classifier ----
    cls_kernel<<<(nA * nC + 255) / 256, 256, 0, stream>>>(h2, aspects, W_cls, b_cls,
                                                          (float*)d_out, nA, nC);
}